// ClusterGCN_46239617909143
// MI455X (gfx1250) — compile-verified
//
#include <hip/hip_runtime.h>

#define N_NODES 100000
#define N_EDGES 1600000
#define IN_C 128
#define HID_C 128
#define OUT_C 64

typedef __attribute__((ext_vector_type(2))) float v2f;
typedef __attribute__((ext_vector_type(8))) float v8f;

__global__ void zero_f32(float* __restrict__ p, int n) {
  int i = blockIdx.x * blockDim.x + threadIdx.x;
  if (i < n) p[i] = 0.0f;
}

__global__ void degree_kernel(const long long* __restrict__ dst, float* __restrict__ deg, int E) {
  int e = blockIdx.x * blockDim.x + threadIdx.x;
  if (e < E) atomicAdd(&deg[(int)dst[e]], 1.0f);
}

__global__ void rdeg_kernel(const float* __restrict__ deg, float* __restrict__ rdeg, int n) {
  int i = blockIdx.x * blockDim.x + threadIdx.x;
  if (i < n) rdeg[i] = 1.0f / fmaxf(deg[i], 1.0f);
}

// One thread per (edge, channel). C must be 128.
template<int C>
__global__ void scatter_add_kernel(const long long* __restrict__ src,
                                   const long long* __restrict__ dst,
                                   const float* __restrict__ feat,
                                   float* __restrict__ sumbuf, int E) {
  static_assert(C == 128, "shift hardcoded for C=128");
  long long idx = (long long)blockIdx.x * blockDim.x + threadIdx.x;
  if (idx >= (long long)E * C) return;
  int e = (int)(idx >> 7);
  int c = (int)(idx & 127);
  int s = (int)src[e];
  int d = (int)dst[e];
  atomicAdd(&sumbuf[(size_t)d * C + c], feat[(size_t)s * C + c]);
}

// Fused SAGE layer: out = act( (suma/deg) @ Wl + bias + xfeat @ Wr )
// One wave computes a 16(nodes) x 16(out-ch) tile via V_WMMA_F32_16X16X4_F32.
// Block: (NOUT/16) waves, each wave handles one N-tile; grid.x covers M tiles.
template<int K, int NOUT, bool RELU>
__global__ void __launch_bounds__(32 * (NOUT / 16))
sage_gemm_kernel(const float* __restrict__ suma, const float* __restrict__ rdeg,
                 const float* __restrict__ xfeat,
                 const float* __restrict__ Wl, const float* __restrict__ Wr,
                 const float* __restrict__ bias, float* __restrict__ out) {
  const int lane = threadIdx.x & 31;
  const int wave = threadIdx.x >> 5;
  const int m0 = blockIdx.x * 16;
  const int n0 = wave * 16;

  // A-matrix f32 16x4 layout: lanes 0-15 -> M=lane, K={0,1}; lanes 16-31 -> K={2,3}
  const int arow = lane & 15;
  const int kb = (lane >> 4) * 2;
  // B-matrix f32 4x16 layout: lanes 0-15 -> N=lane, K={0,1}; lanes 16-31 -> K={2,3}
  const int col = lane & 15;

  const int node = m0 + arow;
  const float rd = rdeg[node];

  const float* aS = suma + (size_t)node * K + kb;
  const float* aX = xfeat + (size_t)node * K + kb;
  const float* bL = Wl + (size_t)kb * NOUT + (n0 + col);
  const float* bR = Wr + (size_t)kb * NOUT + (n0 + col);

  v8f acc = {};
#pragma unroll 4
  for (int k0 = 0; k0 < K; k0 += 4) {
    v2f a1, a2, b1, b2;
    a1.x = aS[k0] * rd;
    a1.y = aS[k0 + 1] * rd;
    a2.x = aX[k0];
    a2.y = aX[k0 + 1];
    b1.x = bL[(size_t)k0 * NOUT];
    b1.y = bL[(size_t)(k0 + 1) * NOUT];
    b2.x = bR[(size_t)k0 * NOUT];
    b2.y = bR[(size_t)(k0 + 1) * NOUT];
    acc = __builtin_amdgcn_wmma_f32_16x16x4_f32(false, a1, false, b1,
                                                (short)0, acc, false, false);
    acc = __builtin_amdgcn_wmma_f32_16x16x4_f32(false, a2, false, b2,
                                                (short)0, acc, false, false);
  }

  // D layout: VGPR r holds M=r (lanes 0-15) and M=r+8 (lanes 16-31), N=lane&15
  const float bv = bias[n0 + col];
  const int rowoff = (lane >> 4) * 8;
#pragma unroll
  for (int r = 0; r < 8; ++r) {
    float v = acc[r] + bv;
    if (RELU) v = fmaxf(v, 0.0f);
    out[(size_t)(m0 + rowoff + r) * NOUT + (n0 + col)] = v;
  }
}

// One wave (32 lanes) per node over 64 channels; shuffle-tree reductions.
__global__ void log_softmax_kernel(float* __restrict__ out, int n_nodes) {
  const int lane = threadIdx.x & 31;
  const int wave = threadIdx.x >> 5;
  const int node = blockIdx.x * (blockDim.x >> 5) + wave;
  if (node >= n_nodes) return;
  float* p = out + (size_t)node * OUT_C;
  float v0 = p[lane];
  float v1 = p[lane + 32];
  float m = fmaxf(v0, v1);
#pragma unroll
  for (int off = 16; off >= 1; off >>= 1) m = fmaxf(m, __shfl_xor(m, off, 32));
  float s = expf(v0 - m) + expf(v1 - m);
#pragma unroll
  for (int off = 16; off >= 1; off >>= 1) s += __shfl_xor(s, off, 32);
  const float ls = logf(s);
  p[lane] = v0 - m - ls;
  p[lane + 32] = v1 - m - ls;
}

extern "C" void kernel_launch(void* const* d_in, const int* in_sizes, int n_in,
                              void* d_out, int out_size, void* d_ws, size_t ws_size,
                              hipStream_t stream) {
  (void)in_sizes; (void)n_in; (void)out_size; (void)ws_size;
  const float* x = (const float*)d_in[0];
  const long long* ei = (const long long*)d_in[1];  // int64 per reference
  const float* Wl1 = (const float*)d_in[2];
  const float* bl1 = (const float*)d_in[3];
  const float* Wr1 = (const float*)d_in[4];
  const float* Wl2 = (const float*)d_in[5];
  const float* bl2 = (const float*)d_in[6];
  const float* Wr2 = (const float*)d_in[7];
  float* out = (float*)d_out;

  const long long* src = ei;            // edge_index[0]
  const long long* dst = ei + N_EDGES;  // edge_index[1]

  float* ws = (float*)d_ws;
  float* deg = ws;                                   // N
  float* rdeg = ws + N_NODES;                        // N
  float* sumb = ws + 2 * (size_t)N_NODES;            // N*128
  float* h = sumb + (size_t)N_NODES * HID_C;         // N*128

  const int T = 256;
  const int nFeat = N_NODES * HID_C;  // 12.8M
  const int scatterBlocks = (int)(((long long)N_EDGES * 128 + T - 1) / T);

  // ---- Layer 1 ----
  zero_f32<<<(N_NODES + T - 1) / T, T, 0, stream>>>(deg, N_NODES);
  zero_f32<<<(nFeat + T - 1) / T, T, 0, stream>>>(sumb, nFeat);
  degree_kernel<<<(N_EDGES + T - 1) / T, T, 0, stream>>>(dst, deg, N_EDGES);
  rdeg_kernel<<<(N_NODES + T - 1) / T, T, 0, stream>>>(deg, rdeg, N_NODES);
  scatter_add_kernel<128><<<scatterBlocks, T, 0, stream>>>(src, dst, x, sumb, N_EDGES);
  sage_gemm_kernel<128, 128, true><<<N_NODES / 16, 256, 0, stream>>>(
      sumb, rdeg, x, Wl1, Wr1, bl1, h);

  // ---- Layer 2 ----
  zero_f32<<<(nFeat + T - 1) / T, T, 0, stream>>>(sumb, nFeat);
  scatter_add_kernel<128><<<scatterBlocks, T, 0, stream>>>(src, dst, h, sumb, N_EDGES);
  sage_gemm_kernel<128, 64, false><<<N_NODES / 16, 128, 0, stream>>>(
      sumb, rdeg, h, Wl2, Wr2, bl2, out);

  // ---- log_softmax over 64 channels ----
  log_softmax_kernel<<<(N_NODES + 7) / 8, 256, 0, stream>>>(out, N_NODES);
}